// ContrastiveEncoderMOE_42786464203499
// MI455X (gfx1250) — compile-verified
//
#include <hip/hip_runtime.h>
#include <math.h>

typedef __attribute__((ext_vector_type(2))) float v2f;
typedef __attribute__((ext_vector_type(8))) float v8f;

#define TLEN   2048
#define CIN    16
#define KW     5
#define KDIM   80      // CIN*KW
#define COCH   128
#define CPG    16      // channels per group (128/8)
#define NEXP   8
#define HIDN   128
#define CTX    64
#define DEMB   16
#define GIN    80

#define XSTR   18                  // t-major x stride (floats), even + bank-friendly
#define XROWS  (TLEN + 4)          // 2 halo rows each side
#define WSTR   82                  // weight LDS row stride (floats), even + bank-friendly
#define XLDS_N (XROWS * XSTR)      // 36936 floats
#define HBUF_N (CPG * TLEN)        // 32768 floats
#define WLDS_N (CPG * WSTR)        // 1312 floats
#define RED_N  256
#define GB_N   32

__device__ __forceinline__ float gelu_erf(float v) {
    return 0.5f * v * (1.0f + erff(v * 0.70710678118654752f));
}

// Stage x[b] (16 x 2048, channel-major in global) into LDS t-major with zero halo:
// xlds[(t+2)*XSTR + cin] = x[cin][t];  rows t+2 in {0,1,2050,2051} = 0.
__device__ __forceinline__ void stage_x(const float* __restrict__ xb, float* xlds) {
    const int tid = threadIdx.x;
    if (tid < 4 * XSTR) {
        const int r  = tid / XSTR;                 // 0..3
        const int rr = tid - r * XSTR;
        const int row = (r < 2) ? r : (2048 + r);  // 0,1,2050,2051
        xlds[row * XSTR + rr] = 0.0f;
    }
    for (int idx = tid; idx < CIN * TLEN; idx += 128) {
        const int cin = idx >> 11;
        const int t   = idx & (TLEN - 1);
        xlds[(t + 2) * XSTR + cin] = xb[idx];
    }
}

// Conv1d (16 out-ch x 2048, K=5 'same') via WMMA f32 16x16x4 from LDS-staged x,
// + bias, GroupNorm over the 16x2048 block, GELU. Result left in hbuf.
// Block = 128 threads (4 waves), dual-tile per wave.
// K-ordering: slot = k*16 + cin  ->  within WMMA j: k = j>>2 (constant),
// cin = (4j & 15) + khi, so A and B are single 8B LDS loads w/ immediate offsets.
__device__ __forceinline__ void conv16_gn_gelu_core(
    const float* __restrict__ wrow,      // 16 x 80 weights (global, [ch][cin*5+k])
    const float* __restrict__ bias16,    // 16 or nullptr
    const float* __restrict__ gamma16,
    const float* __restrict__ beta16,
    const float* xlds, float* hbuf, float* wlds, float* red, float* gb)
{
    const int tid  = threadIdx.x;
    const int lane = tid & 31;
    const int wave = tid >> 5;

    // stage weights permuted to slot order: wlds[ch*WSTR + k*16+cin] = w[ch][cin*5+k]
    for (int i = tid; i < CPG * KDIM; i += 128) {
        const int ch  = i / KDIM;
        const int s   = i - ch * KDIM;
        const int k   = s >> 4;
        const int cin = s & 15;
        wlds[ch * WSTR + s] = wrow[ch * KDIM + cin * KW + k];
    }
    if (tid < CPG) {
        gb[tid]       = gamma16[tid];
        gb[16 + tid]  = beta16[tid];
    }
    __syncthreads();

    const int mlan = lane & 15;          // M (A) / N (B) within half-wave
    const int khi  = (lane >> 4) << 1;   // 0 (lanes 0-15) or 2 (lanes 16-31)
    const int crow = (lane >> 4) << 3;   // C/D: +8 rows for upper half

    const float* aBase = wlds + (mlan * WSTR + khi);

    float lsum = 0.0f, lsq = 0.0f;

    // dual-tile: pair p covers tiles 2p and 2p+1  (t0 = 32p, 32p+16)
    for (int p = wave; p < TLEN / 32; p += 4) {
        const int t0 = p * 32;
        const float* bBase0 = xlds + ((t0 + mlan) * XSTR + khi);
        const float* bBase1 = bBase0 + 16 * XSTR;
        v8f acc0 = {0.f, 0.f, 0.f, 0.f, 0.f, 0.f, 0.f, 0.f};
        v8f acc1 = {0.f, 0.f, 0.f, 0.f, 0.f, 0.f, 0.f, 0.f};
#pragma unroll
        for (int j = 0; j < KDIM / 4; ++j) {
            const int k    = j >> 2;          // compile-time per unrolled j
            const int c0   = (4 * j) & 15;    // compile-time per unrolled j
            const int aoff = 4 * j;           // slot offset in wlds row
            const int boff = k * XSTR + c0;   // dword offset from bBase
            const v2f av  = *(const v2f*)(aBase + aoff);
            const v2f bv0 = *(const v2f*)(bBase0 + boff);
            const v2f bv1 = *(const v2f*)(bBase1 + boff);
            acc0 = __builtin_amdgcn_wmma_f32_16x16x4_f32(
                false, av, false, bv0, (short)0, acc0, false, false);
            acc1 = __builtin_amdgcn_wmma_f32_16x16x4_f32(
                false, av, false, bv1, (short)0, acc1, false, false);
        }
#pragma unroll
        for (int i = 0; i < 8; ++i) {
            const int ch = i + crow;
            const float bv = bias16 ? bias16[ch] : 0.0f;
            float v0 = acc0[i] + bv;
            float v1 = acc1[i] + bv;
            hbuf[ch * TLEN + t0 + mlan]      = v0;
            hbuf[ch * TLEN + t0 + 16 + mlan] = v1;
            lsum += v0 + v1;
            lsq  += v0 * v0 + v1 * v1;
        }
    }

    // block reduction for GroupNorm stats over 16*2048 values
    red[tid]       = lsum;
    red[128 + tid] = lsq;
    __syncthreads();
    for (int s = 64; s > 0; s >>= 1) {
        if (tid < s) { red[tid] += red[tid + s]; red[128 + tid] += red[128 + tid + s]; }
        __syncthreads();
    }
    const float inv_n = 1.0f / (float)(CPG * TLEN);
    const float mu    = red[0] * inv_n;
    const float var   = red[128] * inv_n - mu * mu;
    const float rstd  = rsqrtf(var + 1e-5f);

    // normalize + affine + GELU, in place
    for (int idx = tid; idx < CPG * TLEN; idx += 128) {
        const int ch = idx >> 11;
        float v = (hbuf[idx] - mu) * rstd * gb[ch] + gb[16 + ch];
        hbuf[idx] = gelu_erf(v);
    }
    __syncthreads();
}

// ---------------- Kernel 1: router conv -> GN -> GELU -> GAP ----------------
__global__ __launch_bounds__(128)
void moe_router_conv_kernel(const float* __restrict__ x,
                            const float* __restrict__ rw,
                            const float* __restrict__ rg,
                            const float* __restrict__ rb,
                            float* __restrict__ r0)
{
    const int g = blockIdx.x;   // group 0..7
    const int b = blockIdx.y;   // batch 0..31
    extern __shared__ float sm[];
    float* xlds = sm;
    float* hbuf = xlds + XLDS_N;
    float* wlds = hbuf + HBUF_N;
    float* red  = wlds + WLDS_N;
    float* gb   = red + RED_N;

    stage_x(x + (size_t)b * CIN * TLEN, xlds);
    __syncthreads();

    conv16_gn_gelu_core(rw + g * CPG * KDIM, nullptr, rg + g * CPG, rb + g * CPG,
                        xlds, hbuf, wlds, red, gb);

    // GAP over time: 8 threads per channel
    const int tid = threadIdx.x;
    const int ch  = tid >> 3;
    const int sub = tid & 7;
    float s = 0.0f;
    for (int t = sub * 256; t < (sub + 1) * 256; ++t) s += hbuf[ch * TLEN + t];
    red[tid] = s;
    __syncthreads();
    if (tid < CPG) {
        float tot = 0.0f;
        for (int k = 0; k < 8; ++k) tot += red[tid * 8 + k];
        r0[b * HIDN + g * CPG + tid] = tot * (1.0f / (float)TLEN);
    }
}

// ---------------- Kernel 2: MLPs + gate (tiny) ----------------
__global__ __launch_bounds__(128)
void moe_gate_kernel(const float* __restrict__ r0, const float* __restrict__ demo,
                     const float* __restrict__ m1w, const float* __restrict__ m1b,
                     const float* __restrict__ lng, const float* __restrict__ lnb,
                     const float* __restrict__ m2w, const float* __restrict__ m2b,
                     const float* __restrict__ d1w, const float* __restrict__ d1b,
                     const float* __restrict__ dlng, const float* __restrict__ dlnb,
                     const float* __restrict__ d2w, const float* __restrict__ d2b,
                     const float* __restrict__ gw,  const float* __restrict__ gb,
                     float* __restrict__ wh)
{
    const int b = blockIdx.x;
    const int tid = threadIdx.x;
    __shared__ float rrow[HIDN], h1[HIDN], tmp[HIDN], rvec[GIN], dh[32], stats[2];

    rrow[tid] = r0[b * HIDN + tid];
    __syncthreads();

    float z = m1b[tid];
    for (int k = 0; k < HIDN; ++k) z += m1w[tid * HIDN + k] * rrow[k];
    tmp[tid] = z;
    __syncthreads();
    if (tid == 0) {
        float s = 0.f, q = 0.f;
        for (int k = 0; k < HIDN; ++k) { s += tmp[k]; q += tmp[k] * tmp[k]; }
        const float mu = s / (float)HIDN;
        stats[0] = mu; stats[1] = rsqrtf(q / (float)HIDN - mu * mu + 1e-5f);
    }
    __syncthreads();
    h1[tid] = gelu_erf((z - stats[0]) * stats[1] * lng[tid] + lnb[tid]);
    __syncthreads();

    if (tid < CTX) {
        float z2 = m2b[tid];
        for (int k = 0; k < HIDN; ++k) z2 += m2w[tid * HIDN + k] * h1[k];
        rvec[tid] = z2;
    }
    if (tid < 32) {
        float zd = d1b[tid];
        for (int k = 0; k < 8; ++k) zd += d1w[tid * 8 + k] * demo[b * 8 + k];
        tmp[tid] = zd;
    }
    __syncthreads();
    if (tid == 0) {
        float s = 0.f, q = 0.f;
        for (int k = 0; k < 32; ++k) { s += tmp[k]; q += tmp[k] * tmp[k]; }
        const float mu = s / 32.0f;
        stats[0] = mu; stats[1] = rsqrtf(q / 32.0f - mu * mu + 1e-5f);
    }
    __syncthreads();
    if (tid < 32) dh[tid] = gelu_erf((tmp[tid] - stats[0]) * stats[1] * dlng[tid] + dlnb[tid]);
    __syncthreads();
    if (tid < DEMB) {
        float z3 = d2b[tid];
        for (int k = 0; k < 32; ++k) z3 += d2w[tid * 32 + k] * dh[k];
        rvec[CTX + tid] = z3;
    }
    __syncthreads();

    if (tid == 0) {
        float lg[NEXP], p[NEXP];
        for (int e = 0; e < NEXP; ++e) {
            float z4 = gb[e];
            for (int k = 0; k < GIN; ++k) z4 += gw[e * GIN + k] * rvec[k];
            lg[e] = z4;   // TEMP = 1.0
        }
        float mx = lg[0];
        for (int e = 1; e < NEXP; ++e) mx = fmaxf(mx, lg[e]);
        float se = 0.f;
        for (int e = 0; e < NEXP; ++e) { p[e] = expf(lg[e] - mx); se += p[e]; }
        for (int e = 0; e < NEXP; ++e) p[e] /= se;
        int i1 = 0;
        for (int e = 1; e < NEXP; ++e) if (p[e] > p[i1]) i1 = e;
        int i2 = -1;
        for (int e = 0; e < NEXP; ++e) { if (e == i1) continue; if (i2 < 0 || p[e] > p[i2]) i2 = e; }
        const float s2 = p[i1] + p[i2] + 1e-9f;
        for (int e = 0; e < NEXP; ++e)
            wh[b * NEXP + e] = (e == i1 || e == i2) ? p[e] / s2 : 0.0f;
    }
}

// ---------------- Kernel 3: shared + top-k experts, fused combine ----------------
__global__ __launch_bounds__(128)
void moe_out_kernel(const float* __restrict__ x, const float* __restrict__ wh,
                    const float* __restrict__ ew, const float* __restrict__ eb,
                    const float* __restrict__ eg, const float* __restrict__ ebt,
                    const float* __restrict__ sw, const float* __restrict__ sb,
                    const float* __restrict__ sg, const float* __restrict__ sbt,
                    float* __restrict__ out)
{
    const int g = blockIdx.x;
    const int b = blockIdx.y;
    extern __shared__ float sm[];
    float* xlds = sm;
    float* hbuf = xlds + XLDS_N;
    float* wlds = hbuf + HBUF_N;
    float* red  = wlds + WLDS_N;
    float* gbuf = red + RED_N;

    stage_x(x + (size_t)b * CIN * TLEN, xlds);   // staged once, reused for all sources
    __syncthreads();

    float* outg = out + (size_t)b * COCH * TLEN + (size_t)g * CPG * TLEN;

    for (int src = 0; src <= NEXP; ++src) {
        float w;
        const float *wr, *bi, *ga, *be;
        if (src == 0) {
            w = 1.0f;
            wr = sw + g * CPG * KDIM; bi = sb + g * CPG;
            ga = sg + g * CPG;        be = sbt + g * CPG;
        } else {
            const int e = src - 1;
            w = wh[b * NEXP + e];       // uniform across block
            if (w == 0.0f) continue;    // skip non-top-k experts (mask gives exact 0)
            wr = ew + ((size_t)e * COCH + g * CPG) * KDIM;
            bi = eb + e * COCH + g * CPG;
            ga = eg + e * COCH + g * CPG;
            be = ebt + e * COCH + g * CPG;
        }
        conv16_gn_gelu_core(wr, bi, ga, be, xlds, hbuf, wlds, red, gbuf);

        // this WG exclusively owns its out region -> plain RMW, no atomics
        for (int idx = threadIdx.x; idx < CPG * TLEN; idx += 128) {
            const float v = w * hbuf[idx];
            if (src == 0) outg[idx] = v;
            else          outg[idx] += v;
        }
        __syncthreads();
    }
}

extern "C" void kernel_launch(void* const* d_in, const int* in_sizes, int n_in,
                              void* d_out, int out_size, void* d_ws, size_t ws_size,
                              hipStream_t stream) {
    const float* x    = (const float*)d_in[0];
    const float* demo = (const float*)d_in[1];
    const float* rw   = (const float*)d_in[2];
    const float* rg   = (const float*)d_in[3];
    const float* rb   = (const float*)d_in[4];
    const float* m1w  = (const float*)d_in[5];
    const float* m1b  = (const float*)d_in[6];
    const float* lng  = (const float*)d_in[7];
    const float* lnb  = (const float*)d_in[8];
    const float* m2w  = (const float*)d_in[9];
    const float* m2b  = (const float*)d_in[10];
    const float* d1w  = (const float*)d_in[11];
    const float* d1b  = (const float*)d_in[12];
    const float* dlng = (const float*)d_in[13];
    const float* dlnb = (const float*)d_in[14];
    const float* d2w  = (const float*)d_in[15];
    const float* d2b  = (const float*)d_in[16];
    const float* gw   = (const float*)d_in[17];
    const float* gb   = (const float*)d_in[18];
    const float* ew   = (const float*)d_in[19];
    const float* eb   = (const float*)d_in[20];
    const float* eg   = (const float*)d_in[21];
    const float* ebt  = (const float*)d_in[22];
    const float* sw   = (const float*)d_in[23];
    const float* sb   = (const float*)d_in[24];
    const float* sg   = (const float*)d_in[25];
    const float* sbt  = (const float*)d_in[26];
    float* out = (float*)d_out;

    float* r0 = (float*)d_ws;                 // 32*128 floats
    float* wh = r0 + 32 * HIDN;               // 32*8 floats

    const size_t lds_bytes =
        (size_t)(XLDS_N + HBUF_N + WLDS_N + RED_N + GB_N) * sizeof(float);
    (void)hipFuncSetAttribute((const void*)moe_router_conv_kernel,
                              hipFuncAttributeMaxDynamicSharedMemorySize, (int)lds_bytes);
    (void)hipFuncSetAttribute((const void*)moe_out_kernel,
                              hipFuncAttributeMaxDynamicSharedMemorySize, (int)lds_bytes);

    moe_router_conv_kernel<<<dim3(8, 32), 128, lds_bytes, stream>>>(x, rw, rg, rb, r0);
    moe_gate_kernel<<<32, 128, 0, stream>>>(r0, demo, m1w, m1b, lng, lnb, m2w, m2b,
                                            d1w, d1b, dlng, dlnb, d2w, d2b, gw, gb, wh);
    moe_out_kernel<<<dim3(8, 32), 128, lds_bytes, stream>>>(x, wh, ew, eb, eg, ebt,
                                                            sw, sb, sg, sbt, out);
}